// GCNIILayer_29145648071321
// MI455X (gfx1250) — compile-verified
//
#include <hip/hip_runtime.h>
#include <hip/hip_bf16.h>

// GCNII layer for MI455X (gfx1250, wave32).
//   hidden = segment_sum(edge_attr * x[col], row)       -> f32 atomics into d_out (L2-resident)
//   hidden' = 0.9*hidden + 0.1*init_x
//   out     = 0.5*(hidden' @ W) + 0.5*hidden'           -> V_WMMA_F32_16X16X4_F32, in-place on d_out

#define CH     128
#define ALPHA  0.1f
#define BETA   0.5f
#define SW_ST  132   // padded LDS row stride (floats) for weight chunk
#define SH_ST  132   // padded LDS row stride (floats) for H tile

typedef __attribute__((ext_vector_type(2))) float v2f;
typedef __attribute__((ext_vector_type(8))) float v8f;

// ---------------- kernel 1: zero the accumulator (d_out) ----------------
__global__ void gcnii_zero_kernel(float4* __restrict__ p, long long n4) {
    long long i = (long long)blockIdx.x * blockDim.x + threadIdx.x;
    long long stride = (long long)gridDim.x * blockDim.x;
    for (; i < n4; i += stride) p[i] = float4{0.f, 0.f, 0.f, 0.f};
}

// ---------------- kernel 2: SpMM scatter (edge-parallel) ----------------
// 256 threads = two edges per block; 128 threads cover one edge's channels.
// Gathers hit L2 (x = 51.2MB << 192MB L2); scatter via hardware f32 atomics.
__global__ __launch_bounds__(256) void gcnii_spmm_kernel(
        const float* __restrict__ x,
        const int*   __restrict__ rows,
        const int*   __restrict__ cols,
        const float* __restrict__ eattr,
        float*       __restrict__ hidden,
        int E) {
    int e = blockIdx.x * 2 + (threadIdx.x >> 7);
    int c = threadIdx.x & (CH - 1);
    if (e >= E) return;
    int r  = rows[e];
    int cl = cols[e];
    float w = eattr[e];
    // prefetch the next edge pair's gather row (global_prefetch_b8)
    int en = e + 2;
    if (en < E) __builtin_prefetch(&x[(long long)cols[en] * CH + c], 0, 1);
    float v = w * x[(long long)cl * CH + c];
    unsafeAtomicAdd(&hidden[(long long)r * CH + c], v);   // global_atomic_add_f32
}

// ---------------- kernel 3: fused axpy + GEMM via WMMA, in-place ----------------
// Block = 256 threads = 8 waves. Block owns rows [16*b, 16*b+16).
// Each wave computes one 16x16 output tile (wave id = column tile).
__global__ __launch_bounds__(256) void gcnii_gemm_kernel(
        float*       __restrict__ hidden_out,   // in: hidden sum, out: result (in place)
        const float* __restrict__ init_x,
        const float* __restrict__ weight,       // [CH][CH] row-major (k, n)
        int n_rows) {
    __shared__ float sW[32 * SW_ST];   // one 32-row K-chunk of W (~16.9 KB)
    __shared__ float sH[16 * SH_ST];   // H' tile (~8.4 KB)

    const int tid = threadIdx.x;
    const long long row0 = (long long)blockIdx.x * 16;

    // Stage H' = (1-ALPHA)*hidden + ALPHA*init_x (16 x 128 tile)
    for (int i = tid; i < 16 * CH; i += 256) {
        int rl = i >> 7, c = i & (CH - 1);
        long long r = row0 + rl;
        if (r >= n_rows) r = n_rows - 1;          // N is a multiple of 16; clamp for safety
        float hv = hidden_out[r * CH + c];
        float iv = init_x[r * CH + c];
        sH[rl * SH_ST + c] = (1.0f - ALPHA) * hv + ALPHA * iv;
    }

    const int wave = tid >> 5;          // 0..7 -> output column tile
    const int lane = tid & 31;
    const int m    = lane & 15;         // row within tile (A/B lane row)
    const int hi   = lane >> 4;         // lane half: selects K pair {0,1} vs {2,3}
    const int n    = (wave << 4) + m;   // output column 0..127

    v8f acc = {};

    for (int kc = 0; kc < CH; kc += 32) {
        // Stage weight rows [kc, kc+32)
        for (int i = tid; i < 32 * CH; i += 256) {
            int k = i >> 7, nn = i & (CH - 1);
            sW[k * SW_ST + nn] = weight[(long long)(kc + k) * CH + nn];
        }
        __syncthreads();

        #pragma unroll
        for (int kk = 0; kk < 32; kk += 4) {
            const int ka = kk + 2 * hi;           // K offset of this lane-half's pair
            v2f a, b;
            // A (16x4 f32): lane m holds H'[m][ka], H'[m][ka+1]; fold BETA into A
            a.x = BETA * sH[m * SH_ST + (kc + ka)];
            a.y = BETA * sH[m * SH_ST + (kc + ka) + 1];
            // B (4x16 f32): VGPR i holds W[ka+i][n]
            b.x = sW[ka * SW_ST + n];
            b.y = sW[(ka + 1) * SW_ST + n];
            acc = __builtin_amdgcn_wmma_f32_16x16x4_f32(
                    /*neg_a=*/false, a, /*neg_b=*/false, b,
                    /*c_mod=*/(short)0, acc, /*reuse_a=*/false, /*reuse_b=*/false);
        }
        __syncthreads();   // before next chunk overwrites sW
    }

    // C/D layout: VGPR i -> M = i + 8*hi, N = n. Add (1-BETA)*H' and store in place.
    #pragma unroll
    for (int i = 0; i < 8; ++i) {
        int mm = i + 8 * hi;
        long long r = row0 + mm;
        if (r < n_rows) {
            hidden_out[r * CH + n] = acc[i] + (1.0f - BETA) * sH[mm * SH_ST + n];
        }
    }
}

extern "C" void kernel_launch(void* const* d_in, const int* in_sizes, int n_in,
                              void* d_out, int out_size, void* d_ws, size_t ws_size,
                              hipStream_t stream) {
    const float* x      = (const float*)d_in[0];   // [N, 128]
    const int*   eidx   = (const int*)  d_in[1];   // [2, E] flat: rows then cols
    const float* eattr  = (const float*)d_in[2];   // [E]
    const float* init_x = (const float*)d_in[3];   // [N, 128]
    const float* weight = (const float*)d_in[4];   // [128, 128]
    float*       out    = (float*)d_out;           // [N, 128]

    const int N = in_sizes[0] / CH;
    const int E = in_sizes[1] / 2;

    // 1) zero accumulator (d_out doubles as hidden buffer; re-zeroed every call)
    long long n4 = (long long)N * CH / 4;
    gcnii_zero_kernel<<<2048, 256, 0, stream>>>((float4*)out, n4);

    // 2) edge scatter: 2 edges per 256-thread block
    gcnii_spmm_kernel<<<(E + 1) / 2, 256, 0, stream>>>(x, eidx, eidx + E, eattr, out, E);

    // 3) fused axpy + WMMA GEMM, in place on d_out
    gcnii_gemm_kernel<<<(N + 15) / 16, 256, 0, stream>>>(out, init_x, weight, N);
}